// SpecAugment_48790828482692
// MI455X (gfx1250) — compile-verified
//
#include <hip/hip_runtime.h>

// SpecAugment: [B, N_MELS, T] f32, 2 freq masks + 2 time masks per batch.
// Pure memory-bound elementwise op (~262 MB traffic, ~11 us at 23.3 TB/s HBM).
// Strategy: one 256-thread (8-wave) block per (b, m) row; b128 vector memory
// ops; non-temporal stores to protect the 192 MB L2 (input fits and stays
// resident across replays); skip the input read entirely for freq-masked rows.

constexpr int kB     = 64;
constexpr int kMELS  = 128;
constexpr int kT     = 4000;       // divisible by 4
constexpr int kVecT  = kT / 4;     // 1000 float4's per row

typedef __attribute__((ext_vector_type(4))) float v4f;

__device__ __forceinline__ float mask_sel(float val, int tt,
                                          int ts0, unsigned tw0,
                                          int ts1, unsigned tw1) {
    // tt in [t0, t0+w) for either interval  <=>  (unsigned)(tt - t0) < w
    const bool m = ((unsigned)(tt - ts0) < tw0) | ((unsigned)(tt - ts1) < tw1);
    return m ? 0.0f : val;
}

__global__ __launch_bounds__(256)
void specaugment_kernel(const float* __restrict__ mel,
                        const int* __restrict__ f,    // [2, B]
                        const int* __restrict__ f0,   // [2, B]
                        const int* __restrict__ t,    // [2, B]
                        const int* __restrict__ t0,   // [2, B]
                        float* __restrict__ out) {
    const int row = blockIdx.x;        // 0 .. B*N_MELS-1
    const int b   = row >> 7;          // row / 128
    const int m   = row & (kMELS - 1); // row % 128

    // Block-uniform mask parameters -> scalar loads (s_load_b32).
    const int      fs0 = f0[0 * kB + b];
    const unsigned fw0 = (unsigned)f[0 * kB + b];
    const int      fs1 = f0[1 * kB + b];
    const unsigned fw1 = (unsigned)f[1 * kB + b];
    const bool freq_masked =
        ((unsigned)(m - fs0) < fw0) | ((unsigned)(m - fs1) < fw1);

    const size_t base = (size_t)row * kT;
    v4f* __restrict__ out4 = (v4f*)(out + base);

    if (freq_masked) {
        // Entire row is zero: no input read at all (saves ~15% read BW).
        const v4f z = (v4f)(0.0f);
        for (int i = threadIdx.x; i < kVecT; i += blockDim.x) {
            __builtin_nontemporal_store(z, out4 + i);  // th:TH_STORE_NT
        }
        return;
    }

    const int      ts0 = t0[0 * kB + b];
    const unsigned tw0 = (unsigned)t[0 * kB + b];
    const int      ts1 = t0[1 * kB + b];
    const unsigned tw1 = (unsigned)t[1 * kB + b];

    const v4f* __restrict__ in4 = (const v4f*)(mel + base);

    for (int i = threadIdx.x; i < kVecT; i += blockDim.x) {
        v4f v = in4[i];                 // global_load_b128 (RT -> L2 resident)
        const int tb = i << 2;
        v4f r;
        r.x = mask_sel(v.x, tb + 0, ts0, tw0, ts1, tw1);
        r.y = mask_sel(v.y, tb + 1, ts0, tw0, ts1, tw1);
        r.z = mask_sel(v.z, tb + 2, ts0, tw0, ts1, tw1);
        r.w = mask_sel(v.w, tb + 3, ts0, tw0, ts1, tw1);
        __builtin_nontemporal_store(r, out4 + i);  // global_store_b128 th:NT
    }
}

extern "C" void kernel_launch(void* const* d_in, const int* in_sizes, int n_in,
                              void* d_out, int out_size, void* d_ws, size_t ws_size,
                              hipStream_t stream) {
    (void)in_sizes; (void)n_in; (void)out_size; (void)d_ws; (void)ws_size;

    const float* mel = (const float*)d_in[0];  // [64, 128, 4000] f32
    const int*   f   = (const int*)d_in[1];    // [2, 64]
    const int*   f0  = (const int*)d_in[2];    // [2, 64]
    const int*   t   = (const int*)d_in[3];    // [2, 64]
    const int*   t0  = (const int*)d_in[4];    // [2, 64]
    float*       out = (float*)d_out;          // [64, 128, 4000] f32

    dim3 grid(kB * kMELS);   // 8192 blocks, one per (b, mel) row
    dim3 block(256);         // 8 waves (wave32)
    specaugment_kernel<<<grid, block, 0, stream>>>(mel, f, f0, t, t0, out);
}